// DeepKMeans_87222195847764
// MI455X (gfx1250) — compile-verified
//
#include <hip/hip_runtime.h>

// DKM distance-softmax: dist[k,n] = c2[k] + x2[n] - 2*<c_k, x_n>, then
// out = dist * softmax_over_k(-alpha * dist).  K=128, D=64, N=500000.
// Bandwidth-bound (384 MB @ 23.3 TB/s -> ~16.5 us), so:
//  - fp32 WMMA (V_WMMA_F32_16X16X4_F32): exact vs reference, GEMM is free
//  - centroid A-fragments EXPLICITLY register-resident (v2f a[8][16], 256
//    VGPRs) -> zero steady-state LDS traffic for A (LDS reload would move
//    32KB/tile vs 12KB HBM/tile and throttle below the HBM roofline)
//  - double-buffered B prefetch: next tile's 16 global_load_b64 issued
//    before the current tile's compute, hiding HBM latency at ~2 waves/SIMD.

typedef __attribute__((ext_vector_type(2))) float v2f;
typedef __attribute__((ext_vector_type(8))) float v8f;

#define D_DIM        64
#define K_DIM        128
#define LDS_STRIDE   68        // 64 + 4 pad -> conflict-free fragment reads
#define WAVES_PER_WG 8
#define THREADS      (WAVES_PER_WG * 32)

__device__ __forceinline__
void load_b(v2f b[16], const float* __restrict__ x, int n0, int l16, int hlf)
{
    // lane holds B[kt*4 + 2*hlf + {0,1}][l16] = x[n0+l16][kt*4 + 2*hlf + {0,1}]
    const float* xrow = x + (size_t)(n0 + l16) * D_DIM + hlf * 2;
    #pragma unroll
    for (int kt = 0; kt < 16; ++kt)
        b[kt] = *(const v2f*)(xrow + kt * 4);
}

__device__ __forceinline__
void compute_tile(const v2f (&a)[8][16], const v2f b[16],
                  const float* __restrict__ ldsC2,
                  float* __restrict__ out,
                  int n0, int N, int l16, int hlf, float alpha)
{
    // ---- ||x_n||^2 from fragments already in registers ----
    float x2 = 0.f;
    #pragma unroll
    for (int kt = 0; kt < 16; ++kt)
        x2 += b[kt].x * b[kt].x + b[kt].y * b[kt].y;
    x2 += __shfl_xor(x2, 16, 32);       // combine the two half-wave col sets

    // ---- GEMM: 8 M-tiles x 16 K-steps of v_wmma_f32_16x16x4_f32,
    //      A and B both register-resident -> pure WMMA stream ----
    v8f dist[8];
    #pragma unroll
    for (int mt = 0; mt < 8; ++mt) {
        v8f acc = {};
        #pragma unroll
        for (int kt = 0; kt < 16; ++kt)
            acc = __builtin_amdgcn_wmma_f32_16x16x4_f32(
                    false, a[mt][kt], false, b[kt], (short)0, acc, false, false);
        #pragma unroll
        for (int r = 0; r < 8; ++r) {
            float c2 = ldsC2[mt * 16 + hlf * 8 + r];   // broadcast LDS read
            acc[r] = c2 + x2 - 2.0f * acc[r];
        }
        dist[mt] = acc;
    }

    // ---- per-column min over all K=128 (64 local + XOR-16 partner) ----
    float mn = dist[0][0];
    #pragma unroll
    for (int mt = 0; mt < 8; ++mt)
        #pragma unroll
        for (int r = 0; r < 8; ++r)
            mn = fminf(mn, dist[mt][r]);
    mn = fminf(mn, __shfl_xor(mn, 16, 32));

    // ---- e = exp(-alpha*(dist-min)), S = sum_k e ----
    v8f e[8];
    float S = 0.f;
    #pragma unroll
    for (int mt = 0; mt < 8; ++mt)
        #pragma unroll
        for (int r = 0; r < 8; ++r) {
            float ev = __expf(-alpha * (dist[mt][r] - mn));
            e[mt][r] = ev;
            S += ev;
        }
    S += __shfl_xor(S, 16, 32);
    const float invS = 1.0f / S;

    // ---- out[k*N + n] = dist * e / S  (2 x 64B contiguous per store) ----
    const int n = n0 + l16;
    #pragma unroll
    for (int mt = 0; mt < 8; ++mt)
        #pragma unroll
        for (int r = 0; r < 8; ++r) {
            int k = mt * 16 + hlf * 8 + r;
            out[(size_t)k * N + n] = dist[mt][r] * e[mt][r] * invS;
        }
}

__global__ __launch_bounds__(THREADS)
void dkm_wmma_kernel(const float* __restrict__ x,
                     const float* __restrict__ cent,
                     const int*   __restrict__ alpha_p,
                     float*       __restrict__ out,
                     int N, int n_tiles, int total_waves)
{
    __shared__ float ldsC[K_DIM * LDS_STRIDE];
    __shared__ float ldsC2[K_DIM];

    const int tid  = threadIdx.x;
    const int lane = tid & 31;
    const int wave = tid >> 5;

    // ---- cooperative: centroids -> padded LDS (32 KB payload) ----
    for (int i = tid; i < K_DIM * D_DIM; i += THREADS) {
        int r = i >> 6, c = i & 63;
        ldsC[r * LDS_STRIDE + c] = cent[i];
    }
    __syncthreads();
    if (tid < K_DIM) {                 // c2[k] = ||c_k||^2, once per WG
        float s = 0.f;
        for (int c = 0; c < D_DIM; ++c) {
            float v = ldsC[tid * LDS_STRIDE + c];
            s += v * v;
        }
        ldsC2[tid] = s;
    }
    __syncthreads();

    const float alpha = (float)alpha_p[0];
    const int hlf = lane >> 4;
    const int l16 = lane & 15;

    // ---- A fragments: LDS -> explicit registers, once per wave ----
    // a[mt][kt]: lane holds cent[mt*16 + l16][kt*4 + 2*hlf + {0,1}]
    v2f a[8][16];
    #pragma unroll
    for (int mt = 0; mt < 8; ++mt) {
        const float* arow = &ldsC[(mt * 16 + l16) * LDS_STRIDE + hlf * 2];
        #pragma unroll
        for (int kt = 0; kt < 16; ++kt)
            a[mt][kt] = *(const v2f*)(arow + kt * 4);
    }

    // ---- ping-pong double-buffered grid-stride tile loop ----
    int tile = blockIdx.x * WAVES_PER_WG + wave;
    if (tile >= n_tiles) return;

    v2f b0[16], b1[16];
    load_b(b0, x, tile * 16, l16, hlf);

    while (true) {
        int t1 = tile + total_waves;
        if (t1 < n_tiles) load_b(b1, x, t1 * 16, l16, hlf);       // prefetch
        compute_tile(a, b0, ldsC2, out, tile * 16, N, l16, hlf, alpha);
        if (t1 >= n_tiles) break;

        int t2 = t1 + total_waves;
        if (t2 < n_tiles) load_b(b0, x, t2 * 16, l16, hlf);       // prefetch
        compute_tile(a, b1, ldsC2, out, t1 * 16, N, l16, hlf, alpha);
        if (t2 >= n_tiles) break;

        tile = t2;
    }
}

extern "C" void kernel_launch(void* const* d_in, const int* in_sizes, int n_in,
                              void* d_out, int out_size, void* d_ws, size_t ws_size,
                              hipStream_t stream) {
    const float* x     = (const float*)d_in[0];   // [N, 64] fp32
    const float* cent  = (const float*)d_in[1];   // [128, 64] fp32
    const int*   alpha = (const int*)d_in[2];     // scalar int
    float*       out   = (float*)d_out;           // [128, N] fp32

    const int N = in_sizes[0] / D_DIM;            // 500000
    const int n_tiles = N / 16;                   // 31250

    int grid = 1024;                              // persistent-ish waves
    if (grid * WAVES_PER_WG > n_tiles)
        grid = (n_tiles + WAVES_PER_WG - 1) / WAVES_PER_WG;

    dkm_wmma_kernel<<<grid, THREADS, 0, stream>>>(x, cent, alpha, out,
                                                  N, n_tiles, grid * WAVES_PER_WG);
}